// CandidateRegionGenerator_1425929142286
// MI455X (gfx1250) — compile-verified
//
#include <hip/hip_runtime.h>
#include <hip/hip_bf16.h>

typedef __attribute__((ext_vector_type(2))) float v2f;
typedef __attribute__((ext_vector_type(8))) float v8f;

#define HW_ (128 * 128)

// ---------------------------------------------------------------------------
// conv1: 3->128 channels, 3x3 SAME, direct (K=27 too small for WMMA).
// Each thread computes one pixel for all 128 output channels. Weights in LDS.
// Writes RAW conv output (no bias; bias is a no-op under the following BN).
// ---------------------------------------------------------------------------
__global__ __launch_bounds__(256) void conv1_kernel(
    const float* __restrict__ x, const float* __restrict__ w1,
    float* __restrict__ out) {
  __shared__ float ws[128 * 27];
  for (int e = threadIdx.x; e < 128 * 27; e += 256) ws[e] = w1[e];
  __syncthreads();

  const int pixBlock = blockIdx.x & 63;       // 64 blocks of 256 pixels
  const int img      = blockIdx.x >> 6;
  const int p  = pixBlock * 256 + threadIdx.x;
  const int py = p >> 7;
  const int px = p & 127;
  const float* xi = x + (size_t)img * 3 * HW_;

  float v[27];
#pragma unroll
  for (int c = 0; c < 3; ++c)
#pragma unroll
    for (int dy = 0; dy < 3; ++dy)
#pragma unroll
      for (int dx = 0; dx < 3; ++dx) {
        int gy = py + dy - 1, gx = px + dx - 1;
        v[c * 9 + dy * 3 + dx] =
            (gy >= 0 && gy < 128 && gx >= 0 && gx < 128)
                ? xi[c * HW_ + gy * 128 + gx] : 0.f;
      }

  float* o = out + (size_t)img * 128 * HW_ + p;
  for (int oc = 0; oc < 128; ++oc) {
    float acc = 0.f;
#pragma unroll
    for (int t = 0; t < 27; ++t) acc = fmaf(ws[oc * 27 + t], v[t], acc);
    o[oc * HW_] = acc;
  }
}

// ---------------------------------------------------------------------------
// Per-channel sum / sumsq over (B=16, HW=16384), channel-major layout.
// One block per channel; global_prefetch ahead of the access stream.
// ---------------------------------------------------------------------------
template <int C>
__global__ __launch_bounds__(256) void stats_kernel(
    const float* __restrict__ buf, float* __restrict__ stats) {
  const int c = blockIdx.x;
  float s = 0.f, s2 = 0.f;
  for (int i = threadIdx.x; i < 16 * HW_; i += 256) {
    int n = i >> 14, p = i & 16383;
    size_t idx = ((size_t)n * C + c) * HW_ + p;
    __builtin_prefetch(&buf[idx + 2048], 0, 3);   // global_prefetch_b8
    float v = buf[idx];
    s += v;
    s2 += v * v;
  }
  __shared__ float rs[256], rs2[256];
  rs[threadIdx.x] = s;
  rs2[threadIdx.x] = s2;
  __syncthreads();
  for (int off = 128; off > 0; off >>= 1) {
    if (threadIdx.x < off) {
      rs[threadIdx.x] += rs[threadIdx.x + off];
      rs2[threadIdx.x] += rs2[threadIdx.x + off];
    }
    __syncthreads();
  }
  if (threadIdx.x == 0) {
    stats[c] = rs[0];
    stats[C + c] = rs2[0];
  }
}

__global__ void zero_kernel(float* __restrict__ p, int n) {
  int i = blockIdx.x * blockDim.x + threadIdx.x;
  if (i < n) p[i] = 0.f;
}

// ---------------------------------------------------------------------------
// Stats for point-major (transposed) 32-channel features: feat[img][p][c].
// Lanes map to channels -> coalesced 128B rows. One block per image;
// accumulate into pre-zeroed stats via global atomics.
// ---------------------------------------------------------------------------
__global__ __launch_bounds__(256) void statsT32_kernel(
    const float* __restrict__ featT, float* __restrict__ stats) {
  const int img    = blockIdx.x;
  const int c      = threadIdx.x & 31;
  const int stripe = threadIdx.x >> 5;             // 0..7, 2048 pixels each
  const float* base = featT + (size_t)img * 32 * HW_;
  float s = 0.f, s2 = 0.f;
  for (int p = stripe * 2048; p < (stripe + 1) * 2048; ++p) {
    float v = base[(size_t)p * 32 + c];
    s += v;
    s2 += v * v;
  }
  __shared__ float rs[256], rs2[256];
  rs[threadIdx.x] = s;
  rs2[threadIdx.x] = s2;
  __syncthreads();
  if (stripe == 0) {
    float a = 0.f, b = 0.f;
    for (int k = 0; k < 8; ++k) {
      a += rs[k * 32 + c];
      b += rs2[k * 32 + c];
    }
    atomicAdd(&stats[c], a);
    atomicAdd(&stats[32 + c], b);
  }
}

// scale = g * rsqrt(var+eps), shift = beta - mean*scale  (bias cancels in BN)
__global__ void prep_kernel(const float* __restrict__ stats,
                            const float* __restrict__ g,
                            const float* __restrict__ beta,
                            float* __restrict__ scale,
                            float* __restrict__ shift, int C) {
  int c = threadIdx.x;
  if (c < C) {
    const float inv = 1.0f / (16.0f * 16384.0f);
    float mean = stats[c] * inv;
    float var  = stats[C + c] * inv - mean * mean;
    float sc   = g[c] * rsqrtf(var + 1e-5f);
    scale[c] = sc;
    shift[c] = beta[c] - mean * sc;
  }
}

// ---------------------------------------------------------------------------
// Implicit-GEMM 3x3 SAME conv with V_WMMA_F32_16X16X4_F32.
//   GEMM: M = COUT (16/tile), N = pixels (16/tile), K = CIN*9 (4 per wmma).
// Block = 256 threads = 8 waves; tile layout: (COUT/16) oc-tiles x px-tiles.
// Previous layer's BN+ReLU is applied while staging the input patch into LDS.
// ic-chunk outer / tap inner: each lane's two 9-float weight rows are
// contiguous -> wide b128 weight loads. TR=true stores output point-major
// (feat[p][c]) so the D fragment becomes two contiguous b128 stores.
// ---------------------------------------------------------------------------
template <int CIN, int COUT, int PIX, bool TR>
__global__ __launch_bounds__(256) void conv3x3_bn_wmma(
    const float* __restrict__ in, const float* __restrict__ scale,
    const float* __restrict__ shift, const float* __restrict__ w,
    float* __restrict__ out) {
  constexpr int PC = PIX + 2;             // halo columns
  __shared__ float patch[CIN * 3 * PC];   // [c][ry][col], BN+ReLU applied

  const int tid = threadIdx.x;
  constexpr int BPR = 128 / PIX;          // blocks per row
  const int bx  = blockIdx.x % BPR;
  const int by  = (blockIdx.x / BPR) % 128;
  const int img = blockIdx.x / (BPR * 128);
  const int x0  = bx * PIX;
  const float* inImg = in + (size_t)img * CIN * HW_;

  // ---- stage input patch (3 rows x PC cols x CIN ch) with BN+ReLU ----
  for (int e = tid; e < CIN * 3 * PC; e += 256) {
    int col = e % PC;
    int t   = e / PC;
    int ry  = t % 3;
    int c   = t / 3;
    int gx = x0 + col - 1;
    int gy = by + ry - 1;
    float v = 0.f;
    if (gx >= 0 && gx < 128 && gy >= 0 && gy < 128) {
      float r = inImg[c * HW_ + gy * 128 + gx];
      v = fmaf(r, scale[c], shift[c]);
      v = v > 0.f ? v : 0.f;                    // ReLU
    }
    patch[e] = v;
  }
  __syncthreads();

  // ---- per-wave 16x16 output tile ----
  const int wave = tid >> 5;
  const int lane = tid & 31;
  constexpr int OCT = COUT / 16;
  const int oc0 = (wave % OCT) * 16;
  const int n0  = (wave / OCT) * 16;
  const int m    = lane & 15;     // A: M row / B: N col
  const int hi   = lane >> 4;     // lane half
  const int koff = hi * 2;        // A/B K offset for this half

  v8f acc = {};
  for (int ic = 0; ic < CIN; ic += 4) {
    // contiguous weight rows for this lane's two K indices (OIHW: 9 taps)
    const float* wrow = w + ((size_t)(oc0 + m) * CIN + ic + koff) * 9;
    float wa0[9], wa1[9];
#pragma unroll
    for (int t = 0; t < 9; ++t) {
      wa0[t] = wrow[t];
      wa1[t] = wrow[t + 9];
    }
#pragma unroll
    for (int ky = 0; ky < 3; ++ky) {
#pragma unroll
      for (int kx = 0; kx < 3; ++kx) {
        v2f a, b;
        a.x = wa0[ky * 3 + kx];
        a.y = wa1[ky * 3 + kx];
        int ib = ((ic + koff) * 3 + ky) * PC + n0 + m + kx;
        b.x = patch[ib];
        b.y = patch[ib + 3 * PC];                 // next input channel
        acc = __builtin_amdgcn_wmma_f32_16x16x4_f32(
            false, a, false, b, (short)0, acc, false, false);
      }
    }
  }

  // ---- store C/D: VGPR r -> M = oc, N = pixel (lane%16) ----
  float* outImg = out + (size_t)img * COUT * HW_;
  const int px = by * 128 + x0 + n0 + m;
  const int ocBase = oc0 + hi * 8;
  if (TR) {                                // point-major: 8 contiguous floats
    float* o = outImg + (size_t)px * COUT + ocBase;
#pragma unroll
    for (int r = 0; r < 8; ++r) o[r] = acc[r];
  } else {                                 // channel-major (NCHW)
#pragma unroll
    for (int r = 0; r < 8; ++r) outImg[(ocBase + r) * HW_ + px] = acc[r];
  }
}

// ---------------------------------------------------------------------------
// K-means: one 1024-thread block per image, features point-major (featT[p][c]).
// Distances via WMMA f32 16x16x4 chained over K=32 (8 steps): C-layout gives
// each low lane all 5 cluster dots for one point -> per-lane argmin. A lane's
// two dims per chunk are adjacent in memory -> one b64 load per chunk; the 16
// point-lines are reused from L0 across chunks. Cluster sums: per-wave
// privatized ds_add_f32, summed serially over waves. Final pass writes regions (channel-
// major, NCHW) to d_out.
// ---------------------------------------------------------------------------
__global__ __launch_bounds__(1024) void kmeans_kernel(
    const float* __restrict__ featT,     // (B,16384,32) raw conv3, transposed
    float* __restrict__ out,             // (B,32,128,128) regions
    const float* __restrict__ scale, const float* __restrict__ shift) {
  __shared__ float centp[16 * 32];       // padded centroids (rows 5..15 = 0)
  __shared__ float cnorm[5];
  __shared__ float sums[32 * 5 * 33];    // per-wave [k][0..31]=sum, [32]=cnt
  __shared__ float red[5 * 33];
  __shared__ float means[5 * 32];

  const int tid  = threadIdx.x;
  const int wave = tid >> 5;
  const int lane = tid & 31;
  const int col  = lane & 15;
  const int hi   = lane >> 4;
  const int koff = hi * 2;
  const float* featImg = featT + (size_t)blockIdx.x * 32 * HW_;
  float* outImg = out + (size_t)blockIdx.x * 32 * HW_;

  // per-lane BN constants for the 16 dims this lane handles (loop-invariant)
  float lsc[16], lsh[16];
#pragma unroll
  for (int t = 0; t < 16; ++t) {
    int c = 4 * (t >> 1) + koff + (t & 1);
    lsc[t] = scale[c];
    lsh[t] = shift[c];
  }

  if (tid < 512) {                       // init centroids = BN(first 5 points)
    int k = tid >> 5, c = tid & 31;
    float v = 0.f;
    if (k < 5) v = fmaf(featImg[k * 32 + c], scale[c], shift[c]);
    centp[tid] = v;
  }
  __syncthreads();

  for (int it = 0; it <= 100; ++it) {
    // zero partial sums + centroid norms
    for (int e = tid; e < 32 * 165; e += 1024) sums[e] = 0.f;
    if (tid < 5) {
      float s = 0.f;
      for (int c = 0; c < 32; ++c) {
        float cv = centp[tid * 32 + c];
        s = fmaf(cv, cv, s);
      }
      cnorm[tid] = s;
    }
    __syncthreads();

    // hoist this iteration's A-fragments and cluster norms to registers
    v2f afr[8];
    float cn[5];
#pragma unroll
    for (int ch = 0; ch < 8; ++ch) {
      afr[ch].x = centp[col * 32 + 4 * ch + koff];
      afr[ch].y = centp[col * 32 + 4 * ch + koff + 1];
    }
#pragma unroll
    for (int k = 0; k < 5; ++k) cn[k] = cnorm[k];

    // label + accumulate: each wave owns 512 points = 32 groups of 16
    for (int g = 0; g < 32; ++g) {
      const int p = (wave * 32 + g) * 16 + col;
      const float* pp = featImg + (size_t)p * 32;
      float xv[16];
      v8f acc = {};
#pragma unroll
      for (int ch = 0; ch < 8; ++ch) {
        int c0 = 4 * ch + koff;
        v2f xr = *(const v2f*)(pp + c0);            // aligned b64 load
        float x0v = fmaf(xr.x, lsc[2 * ch], lsh[2 * ch]);
        float x1v = fmaf(xr.y, lsc[2 * ch + 1], lsh[2 * ch + 1]);
        xv[2 * ch] = x0v;
        xv[2 * ch + 1] = x1v;
        v2f b;
        b.x = x0v;
        b.y = x1v;
        acc = __builtin_amdgcn_wmma_f32_16x16x4_f32(
            false, afr[ch], false, b, (short)0, acc, false, false);
      }
      // argmin over 5 clusters (low lanes hold valid dots)
      float best = cn[0] - 2.0f * acc[0];
      int lab = 0;
#pragma unroll
      for (int k = 1; k < 5; ++k) {
        float s = cn[k] - 2.0f * acc[k];
        if (s < best) { best = s; lab = k; }
      }
      lab = __shfl(lab, col);            // broadcast to the high-lane partner
      float* sw = sums + wave * 165 + lab * 33;
#pragma unroll
      for (int t = 0; t < 16; ++t) {
        int c = 4 * (t >> 1) + koff + (t & 1);
        atomicAdd(&sw[c], xv[t]);
      }
      if (hi == 0) atomicAdd(&sw[32], 1.0f);
    }
    __syncthreads();

    // reduce per-wave partials
    if (tid < 165) {
      float t = 0.f;
      for (int w = 0; w < 32; ++w) t += sums[w * 165 + tid];
      red[tid] = t;
    }
    __syncthreads();

    if (it < 100) {                      // centroid update (where cnt > 0)
      if (tid < 160) {
        int k = tid >> 5, c = tid & 31;
        float cnt = red[k * 33 + 32];
        if (cnt > 0.f) centp[k * 32 + c] = red[k * 33 + c] / cnt;
      }
    } else {                             // final means = sums / max(cnt,1)
      if (tid < 160) {
        int k = tid >> 5, c = tid & 31;
        means[tid] = red[k * 33 + c] / fmaxf(red[k * 33 + 32], 1.0f);
      }
    }
    __syncthreads();
  }

  // final assignment: relabel with final centroids, write means[label] to
  // d_out in NCHW layout
  v2f afr[8];
  float cn[5];
#pragma unroll
  for (int ch = 0; ch < 8; ++ch) {
    afr[ch].x = centp[col * 32 + 4 * ch + koff];
    afr[ch].y = centp[col * 32 + 4 * ch + koff + 1];
  }
#pragma unroll
  for (int k = 0; k < 5; ++k) cn[k] = cnorm[k];

  for (int g = 0; g < 32; ++g) {
    const int p = (wave * 32 + g) * 16 + col;
    const float* pp = featImg + (size_t)p * 32;
    v8f acc = {};
#pragma unroll
    for (int ch = 0; ch < 8; ++ch) {
      int c0 = 4 * ch + koff;
      v2f xr = *(const v2f*)(pp + c0);
      float x0v = fmaf(xr.x, lsc[2 * ch], lsh[2 * ch]);
      float x1v = fmaf(xr.y, lsc[2 * ch + 1], lsh[2 * ch + 1]);
      v2f b;
      b.x = x0v;
      b.y = x1v;
      acc = __builtin_amdgcn_wmma_f32_16x16x4_f32(
          false, afr[ch], false, b, (short)0, acc, false, false);
    }
    float best = cn[0] - 2.0f * acc[0];
    int lab = 0;
#pragma unroll
    for (int k = 1; k < 5; ++k) {
      float s = cn[k] - 2.0f * acc[k];
      if (s < best) { best = s; lab = k; }
    }
    lab = __shfl(lab, col);
#pragma unroll
    for (int t = 0; t < 16; ++t) {
      int c = 4 * (t >> 1) + koff + (t & 1);
      outImg[c * HW_ + p] = means[lab * 32 + c];
    }
  }
}

// ---------------------------------------------------------------------------
extern "C" void kernel_launch(void* const* d_in, const int* in_sizes, int n_in,
                              void* d_out, int out_size, void* d_ws,
                              size_t ws_size, hipStream_t stream) {
  const float* x     = (const float*)d_in[0];
  const float* w1    = (const float*)d_in[1];
  const float* g1    = (const float*)d_in[3];
  const float* beta1 = (const float*)d_in[4];
  const float* w2    = (const float*)d_in[5];
  const float* g2    = (const float*)d_in[7];
  const float* beta2 = (const float*)d_in[8];
  const float* w3    = (const float*)d_in[9];
  const float* g3    = (const float*)d_in[11];
  const float* beta3 = (const float*)d_in[12];
  float* out = (float*)d_out;

  float* ws     = (float*)d_ws;
  float* buf1   = ws;                         // 16*128*16384 f32 (134 MB)
  float* buf2   = buf1 + (size_t)16 * 128 * HW_;  // 16*64*16384 f32 (67 MB)
  float* stat   = buf2 + (size_t)16 * 64 * HW_;   // 256 f32 (reused per layer)
  float* scale1 = stat + 256;
  float* shift1 = scale1 + 128;
  float* scale2 = shift1 + 128;
  float* shift2 = scale2 + 64;
  float* scale3 = shift2 + 64;
  float* shift3 = scale3 + 32;
  float* featT  = buf1;   // conv3 output (transposed) reuses dead conv1 buffer

  // layer 1: direct conv, stats, BN prep
  conv1_kernel<<<16 * 64, 256, 0, stream>>>(x, w1, buf1);
  stats_kernel<128><<<128, 256, 0, stream>>>(buf1, stat);
  prep_kernel<<<1, 128, 0, stream>>>(stat, g1, beta1, scale1, shift1, 128);

  // layer 2: WMMA implicit GEMM (BN1+ReLU fused on input)
  conv3x3_bn_wmma<128, 64, 32, false>
      <<<16 * 128 * 4, 256, 0, stream>>>(buf1, scale1, shift1, w2, buf2);
  stats_kernel<64><<<64, 256, 0, stream>>>(buf2, stat);
  prep_kernel<<<1, 64, 0, stream>>>(stat, g2, beta2, scale2, shift2, 64);

  // layer 3: WMMA implicit GEMM -> featT (point-major, overwrites dead buf1)
  conv3x3_bn_wmma<64, 32, 64, true>
      <<<16 * 128 * 2, 256, 0, stream>>>(buf2, scale2, shift2, w3, featT);
  zero_kernel<<<1, 64, 0, stream>>>(stat, 64);
  statsT32_kernel<<<16, 256, 0, stream>>>(featT, stat);
  prep_kernel<<<1, 32, 0, stream>>>(stat, g3, beta3, scale3, shift3, 32);

  // k-means (BN3 fused on read) -> regions written NCHW to d_out
  kmeans_kernel<<<16, 1024, 0, stream>>>(featT, out, scale3, shift3);
}